// JointAngleLoss_51677046505816
// MI455X (gfx1250) — compile-verified
//
#include <hip/hip_runtime.h>
#include <stdint.h>

// JointAngleLoss on MI455X (gfx1250):
//  - B=524288 batches x 21 joints x 3 f32 = 132 MB in, 4 B out -> pure HBM-bound (~5.7us @ 23.3TB/s)
//  - CDNA5 path: GLOBAL_LOAD_ASYNC_TO_LDS_B128 in GVS form (saddr + voffset + imm offset,
//    INST_OFFSET advances BOTH the LDS dest and the global src per CDNA5 ISA),
//    tracked by ASYNCcnt -> 16-deep async pipeline with near-zero address VALU.
//  - conflict-free LDS compute (lane stride 63 dwords, gcd(63,64)=1)
//  - deterministic two-pass reduction (no float atomics)

#define THREADS          256
#define BATCH_PER_BLOCK  256
#define FLOATS_PER_BATCH 63                                        // 21 joints * 3
#define VEC4_PER_BLOCK   (BATCH_PER_BLOCK * FLOATS_PER_BATCH / 4)  // 4032 b128 packets / block

__global__ __launch_bounds__(THREADS)
void hand_loss_pass1(const float* __restrict__ in, float* __restrict__ partial) {
    __shared__ float smem[BATCH_PER_BLOCK * FLOATS_PER_BATCH]; // 64512 B staging
    __shared__ float red[THREADS];                             // 1024 B reduction

    const int tid = threadIdx.x;
    const int bid = blockIdx.x;

    // Block base: bid * 64512 bytes -> 16-byte aligned (64512 % 16 == 0).
    const uint64_t gbase64 = (uint64_t)(uintptr_t)(in + (size_t)bid * (BATCH_PER_BLOCK * FLOATS_PER_BATCH));

    // GVS async copy: mem = SADDR + VADDR32 + offset ; lds = VDST + offset.
    // One LDS-addr VGPR + one 32-bit voffset VGPR serve all 16 packets.
    const uint32_t la   = (uint32_t)(uintptr_t)&smem[0] + (uint32_t)tid * 16u;
    const uint32_t voff = (uint32_t)tid * 16u;

#define ASYNC_LD(OFF)                                                        \
    asm volatile("global_load_async_to_lds_b128 %0, %1, %2 offset:" #OFF     \
                 :: "v"(la), "v"(voff), "s"(gbase64) : "memory")

    ASYNC_LD(0);
    ASYNC_LD(4096);
    ASYNC_LD(8192);
    ASYNC_LD(12288);
    ASYNC_LD(16384);
    ASYNC_LD(20480);
    ASYNC_LD(24576);
    ASYNC_LD(28672);
    ASYNC_LD(32768);
    ASYNC_LD(36864);
    ASYNC_LD(40960);
    ASYNC_LD(45056);
    ASYNC_LD(49152);
    ASYNC_LD(53248);
    ASYNC_LD(57344);
    if (tid < (VEC4_PER_BLOCK - 15 * THREADS)) {  // tail: packets 3840..4031, EXEC-masked
        ASYNC_LD(61440);
    }
#undef ASYNC_LD

    asm volatile("s_wait_asynccnt 0x0" ::: "memory");
    __syncthreads();

    // ---- compute: thread t owns batch t of this block ----------------------
    const int jb = tid * FLOATS_PER_BATCH;
    float acc = 0.0f;

    #pragma unroll
    for (int f = 0; f < 5; ++f) {
        const int o = jb + f * 12;  // joint 4f
        float j0x = smem[o +  0], j0y = smem[o +  1], j0z = smem[o +  2];
        float j1x = smem[o +  3], j1y = smem[o +  4], j1z = smem[o +  5];
        float j2x = smem[o +  6], j2y = smem[o +  7], j2z = smem[o +  8];
        float j3x = smem[o +  9], j3y = smem[o + 10], j3z = smem[o + 11];
        float j4x = smem[o + 12], j4y = smem[o + 13], j4z = smem[o + 14];

        float b1x = j1x - j0x, b1y = j1y - j0y, b1z = j1z - j0z;
        float b2x = j2x - j1x, b2y = j2y - j1y, b2z = j2z - j1z;
        float b3x = j3x - j2x, b3y = j3y - j2y, b3z = j3z - j2z;
        float b4x = j4x - j3x, b4y = j4y - j3y, b4z = j4z - j3z;

        // rot_palm = cross(b2, b1)
        float rpx = b2y * b1z - b2z * b1y;
        float rpy = b2z * b1x - b2x * b1z;
        float rpz = b2x * b1y - b2y * b1x;
        // rot_mid = cross(b3, b2)
        float rmx = b3y * b2z - b3z * b2y;
        float rmy = b3z * b2x - b3x * b2z;
        float rmz = b3x * b2y - b3y * b2x;
        // rot_tip = cross(b4, b3)
        float rtx = b4y * b3z - b4z * b3y;
        float rty = b4z * b3x - b4x * b3z;
        float rtz = b4x * b3y - b4y * b3x;

        float coplane = (rpx + rmx) * b4x + (rpy + rmy) * b4y + (rpz + rmz) * b4z;
        float vd1 = rtx * rmx + rty * rmy + rtz * rmz;
        float vd2 = rpx * rmx + rpy * rmy + rpz * rmz;

        acc += coplane;
        if (vd1 < 0.0f) acc += vd1 * vd1;
        if (vd2 < 0.0f) acc += vd2 * vd2;
    }

    // ---- deterministic in-block tree reduction -----------------------------
    red[tid] = acc;
    __syncthreads();
    #pragma unroll
    for (int s = THREADS / 2; s > 0; s >>= 1) {
        if (tid < s) red[tid] += red[tid + s];
        __syncthreads();
    }
    if (tid == 0) partial[bid] = red[0];
}

__global__ __launch_bounds__(256)
void hand_loss_pass2(const float* __restrict__ partial, float* __restrict__ out, int n) {
    __shared__ float red[256];
    const int t = threadIdx.x;
    float a = 0.0f;
    for (int i = t; i < n; i += 256) a += partial[i];  // fixed order -> deterministic
    red[t] = a;
    __syncthreads();
    #pragma unroll
    for (int s = 128; s > 0; s >>= 1) {
        if (t < s) red[t] += red[t + s];
        __syncthreads();
    }
    if (t == 0) out[0] = red[0];
}

extern "C" void kernel_launch(void* const* d_in, const int* in_sizes, int n_in,
                              void* d_out, int out_size, void* d_ws, size_t ws_size,
                              hipStream_t stream) {
    (void)n_in; (void)out_size; (void)ws_size;
    const float* in  = (const float*)d_in[0];
    float*       out = (float*)d_out;
    float*       ws  = (float*)d_ws;

    const int total   = in_sizes[0];                  // B * 63
    const int B       = total / FLOATS_PER_BATCH;     // 524288
    const int nblocks = B / BATCH_PER_BLOCK;          // 2048

    hand_loss_pass1<<<nblocks, THREADS, 0, stream>>>(in, ws);
    hand_loss_pass2<<<1, 256, 0, stream>>>(ws, out, nblocks);
}